// Symmetrizer_Tensor_2843268350085
// MI455X (gfx1250) — compile-verified
//
#include <hip/hip_runtime.h>

typedef __attribute__((ext_vector_type(2))) float v2f;
typedef __attribute__((ext_vector_type(8))) float v8f;

#define NL 20      // number of (lx,ly,lz) multi-indices for MAX_L=3
#define NC 16      // channels
#define NA2 3      // S2 output channels
#define NA3 13     // S3 keys
#define NOUT 17    // 1 + 3 + 13
#define NPAIR 210  // symmetric (l<=m) pairs of 20
#define KPAD 212   // NPAIR padded to multiple of 4 (WMMA K step)
#define NQ 260     // NL * NA3 (n,a) columns
#define NQPAD 272  // padded to multiple of 16 (WMMA N tile)
#define NTILE 17   // NQPAD / 16

__device__ __forceinline__ void pair_from_p(int p, int& l, int& m) {
  int base = 0; l = 0;
  while (base + (NL - l) <= p) { base += (NL - l); ++l; }
  m = l + (p - base);
}

// Build Wt[q][p] = S3[l,m,n,a] (+ S3[m,l,n,a] if l<m), transposed so that the
// per-lane B-fragment pair (rows kk+koff, kk+koff+1 at a fixed column) is one
// contiguous 8-byte load. Also emits the pair index table. Wt is 230 KB ->
// L2-resident and reused by all 8192 blocks.
__global__ void prep_kernel(const float* __restrict__ S3, float* __restrict__ Wt,
                            int* __restrict__ pairLM) {
  const int idx = blockIdx.x * blockDim.x + threadIdx.x;
  if (idx < NPAIR) {
    int l, m; pair_from_p(idx, l, m);
    pairLM[idx] = l | (m << 8);
  }
  const int total = NQPAD * KPAD;
  for (int i = idx; i < total; i += gridDim.x * blockDim.x) {
    const int q = i / KPAD, p = i % KPAD;
    float v = 0.f;
    if (q < NQ && p < NPAIR) {
      int l, m; pair_from_p(p, l, m);
      const int n = q / NA3, a = q % NA3;
      v = S3[((l * NL + m) * NL + n) * NA3 + a];
      if (l != m) v += S3[((m * NL + l) * NL + n) * NA3 + a];
    }
    Wt[i] = v;
  }
}

// One block per (i,j). 128 threads = 4 wave32 waves.
__global__ void __launch_bounds__(128) sym_main(
    const float* __restrict__ node, const float* __restrict__ S2,
    const float* __restrict__ Wt, const int* __restrict__ pairLM,
    float* __restrict__ out) {
  __shared__ float Vl[NL][NC];       // 1.25 KB
  __shared__ float Pl[NC][KPAD];     // 13.25 KB : A-matrix (k x pair)
  __shared__ float Tl[NC][NQPAD];    // 17 KB    : tmp (k x (n,a))
  const int tid = threadIdx.x;
  const long ij = blockIdx.x;

  // Stage 0: load V (20x16) for this (i,j)
  for (int i = tid; i < NL * NC; i += 128)
    Vl[i / NC][i % NC] = node[ij * (NL * NC) + i];
  __syncthreads();

  // Stage 1: build P[k][p] = V[l,k] * V[m,k] for symmetric pairs; pad with 0
  for (int i = tid; i < NC * KPAD; i += 128) {
    const int k = i / KPAD, p = i % KPAD;
    float v = 0.f;
    if (p < NPAIR) {
      const int lm = pairLM[p];
      v = Vl[lm & 0xff][k] * Vl[lm >> 8][k];
    }
    Pl[k][p] = v;
  }

  // out0 and out2 (S2 is diagonal; uniform scalar loads of the coefficients)
  if (tid < NC) {
    const int k = tid;
    float* o = out + ij * (NOUT * NC);
    o[k] = Vl[0][k];
    for (int a = 0; a < NA2; ++a) {
      float s = 0.f;
      for (int l = 0; l < NL; ++l) {
        const float c = S2[(l * NL + l) * NA2 + a];
        s += c * Vl[l][k] * Vl[l][k];
      }
      o[(1 + a) * NC + k] = s;
    }
  }
  __syncthreads();

  // Stage 2: tmp(16 x 272) = P(16 x 212) @ W(212 x 272) with
  // V_WMMA_F32_16X16X4_F32. Tile/K loops are wave-uniform; EXEC all-1s.
  const int wave = tid >> 5, lane = tid & 31;
  const int M = lane & 15;              // A row / D column index
  const int half = lane >> 4;           // 0: K 0..1 / rows 0..7, 1: K 2..3 / rows 8..15
  const int koff = half << 1;
  for (int t = wave; t < NTILE; t += 4) {
    const int n0 = t * 16;
    const int col = n0 + M;
    const float* wrow = Wt + (long)col * KPAD;  // B column 'col', contiguous in K
    v8f acc = {};
    for (int kk = 0; kk < KPAD; kk += 4) {
      v2f a, b;
      a.x = Pl[M][kk + koff];
      a.y = Pl[M][kk + koff + 1];
      b.x = wrow[kk + koff];
      b.y = wrow[kk + koff + 1];
      // (neg_a, A, neg_b, B, c_mod, C, reuse_a, reuse_b)
      acc = __builtin_amdgcn_wmma_f32_16x16x4_f32(
          false, a, false, b, (short)0, acc, false, false);
    }
#pragma unroll
    for (int r = 0; r < 8; ++r)
      Tl[r + (half << 3)][col] = acc[r];   // D: VGPR r -> row r (+8 for high half)
  }
  __syncthreads();

  // Stage 3: out3[a,k] = sum_n tmp[k, n*13+a] * V[n,k]
  for (int idx = tid; idx < NA3 * NC; idx += 128) {
    const int a = idx / NC, k = idx % NC;
    float s = 0.f;
    for (int n = 0; n < NL; ++n)
      s += Tl[k][n * NA3 + a] * Vl[n][k];
    out[ij * (NOUT * NC) + (4 + a) * NC + k] = s;
  }
}

extern "C" void kernel_launch(void* const* d_in, const int* in_sizes, int n_in,
                              void* d_out, int out_size, void* d_ws, size_t ws_size,
                              hipStream_t stream) {
  const float* node = (const float*)d_in[0];
  const float* S2   = (const float*)d_in[1];
  const float* S3   = (const float*)d_in[2];
  float* out = (float*)d_out;

  float* Wt     = (float*)d_ws;                                        // 272*212 f32
  int*   pairLM = (int*)((char*)d_ws + (size_t)NQPAD * KPAD * sizeof(float));

  const int NR = in_sizes[0] / (NL * NC);   // 8192 (i,j) blocks

  prep_kernel<<<(NQPAD * KPAD + 255) / 256, 256, 0, stream>>>(S3, Wt, pairLM);
  sym_main<<<NR, 128, 0, stream>>>(node, S2, Wt, pairLM, out);
}